// HistogramLoss_7447473292149
// MI455X (gfx1250) — compile-verified
//
#include <hip/hip_runtime.h>

// HistogramLoss for MI455X (gfx1250, wave32).
//
// Kernel 1 (hist_kernel): one workgroup per slice (96 slices of 1 MB).
//   Phase A: streaming float4 loads -> per-slice min/max (wave32 shuffle
//            reduction + LDS combine).
//   Phase B: re-read the SAME 1 MB (L2-resident: 96 MB in flight < 192 MB L2)
//            and bin into per-wave privatized LDS histograms via ds_add_u32.
//   Reduce 32 wave histograms, write normalized hist (count/256) to workspace.
//
// Kernel 2 (loss_kernel): MSE over 48x256 = 12288 diffs using
//   V_WMMA_F32_16X16X4_F32 as a dot-product engine: D = ones(16x4) x B(4x16) + C,
//   B packed with squared diffs, C accumulated across iterations.

#define SLICE_PIX (512 * 512)
#define BINS      256
#define NSLICE    48            // N*C per tensor
#define THREADS   1024
#define WAVESZ    32
#define NWAVE     (THREADS / WAVESZ)

typedef __attribute__((ext_vector_type(2))) float v2f;
typedef __attribute__((ext_vector_type(8))) float v8f;

__global__ __launch_bounds__(THREADS) void hist_kernel(const float* __restrict__ src,
                                                       const float* __restrict__ tgt,
                                                       float* __restrict__ hist_out) {
    __shared__ unsigned int lds_hist[NWAVE * BINS];   // 32 KB, per-wave privatized
    __shared__ float wmin[NWAVE];
    __shared__ float wmax[NWAVE];

    const int g = blockIdx.x;  // 0..95: [0,48) source, [48,96) target
    const float* base = (g < NSLICE) ? (src + (size_t)g * SLICE_PIX)
                                     : (tgt + (size_t)(g - NSLICE) * SLICE_PIX);

    const int tid  = threadIdx.x;
    const int lane = tid & (WAVESZ - 1);
    const int wave = tid >> 5;

    // Zero per-wave histograms (covered by the phase-A barrier).
    for (int i = tid; i < NWAVE * BINS; i += THREADS) lds_hist[i] = 0u;

    // ---- Phase A: per-slice min/max, streaming b128 loads ----
    const float4* p4 = (const float4*)base;
    float lmin =  3.402823466e38f;
    float lmax = -3.402823466e38f;
    for (int i = tid; i < SLICE_PIX / 4; i += THREADS) {
        float4 v = p4[i];
        lmin = fminf(lmin, fminf(fminf(v.x, v.y), fminf(v.z, v.w)));
        lmax = fmaxf(lmax, fmaxf(fmaxf(v.x, v.y), fmaxf(v.z, v.w)));
    }
    // wave32 butterfly reduction
    for (int off = 16; off > 0; off >>= 1) {
        lmin = fminf(lmin, __shfl_xor(lmin, off, WAVESZ));
        lmax = fmaxf(lmax, __shfl_xor(lmax, off, WAVESZ));
    }
    if (lane == 0) { wmin[wave] = lmin; wmax[wave] = lmax; }
    __syncthreads();

    float mn = wmin[0], mx = wmax[0];
    #pragma unroll
    for (int w = 1; w < NWAVE; ++w) {
        mn = fminf(mn, wmin[w]);
        mx = fmaxf(mx, wmax[w]);
    }
    const float width = mx - mn;
    const float scale = (float)BINS / ((width > 0.0f) ? width : 1.0f);

    // ---- Phase B: re-read (L2-hot) and bin into this wave's LDS histogram ----
    unsigned int* myHist = &lds_hist[wave * BINS];
    for (int i = tid; i < SLICE_PIX / 4; i += THREADS) {
        float4 v = p4[i];
        int b0 = min(max((int)((v.x - mn) * scale), 0), BINS - 1);
        int b1 = min(max((int)((v.y - mn) * scale), 0), BINS - 1);
        int b2 = min(max((int)((v.z - mn) * scale), 0), BINS - 1);
        int b3 = min(max((int)((v.w - mn) * scale), 0), BINS - 1);
        atomicAdd(&myHist[b0], 1u);   // ds_add_u32
        atomicAdd(&myHist[b1], 1u);
        atomicAdd(&myHist[b2], 1u);
        atomicAdd(&myHist[b3], 1u);
    }
    __syncthreads();

    // ---- Reduce 32 wave histograms; write normalized hist (count / bins) ----
    for (int b = tid; b < BINS; b += THREADS) {
        unsigned int s = 0;
        #pragma unroll
        for (int w = 0; w < NWAVE; ++w) s += lds_hist[w * BINS + b];
        hist_out[(size_t)g * BINS + b] = (float)s * (1.0f / (float)BINS);
    }
}

// ---- Loss: mean((hs - ht)^2) over 48*256 elements via WMMA f32 16x16x4 ----
// Each wave per iteration packs 64 squared diffs into B (v2f per lane) and does
// D = ones(16x4) x B(4x16) + C. Every D element equals a column sum of B, so
// the total over one D row (lanes 0..15, element 0) is the sum of all 64 values.
__global__ __launch_bounds__(256) void loss_kernel(const float* __restrict__ hist,
                                                   float* __restrict__ out) {
    __shared__ float psum[8];
    const int tid  = threadIdx.x;
    const int lane = tid & 31;
    const int wave = tid >> 5;

    const float* hs = hist;
    const float* ht = hist + NSLICE * BINS;

    v2f a; a[0] = 1.0f; a[1] = 1.0f;   // all-ones A (16x4)
    v8f c = {};                         // accumulator C/D (16x16 f32)

    const int nChunks = (NSLICE * BINS) / 64;   // 192 chunks of 64 elements
    for (int ch = wave; ch < nChunks; ch += 8) {
        const int i0 = ch * 64 + lane * 2;
        float d0 = hs[i0]     - ht[i0];
        float d1 = hs[i0 + 1] - ht[i0 + 1];
        v2f b; b[0] = d0 * d0; b[1] = d1 * d1;
        // 8 args: (neg_a, A, neg_b, B, c_mod, C, reuse_a, reuse_b)
        c = __builtin_amdgcn_wmma_f32_16x16x4_f32(
                false, a, false, b, (short)0, c, false, false);
    }

    // Row 0 of D lives in element 0 of lanes 0..15; lanes 16..31 duplicate it.
    float part = (lane < 16) ? c[0] : 0.0f;
    for (int off = 16; off > 0; off >>= 1) part += __shfl_xor(part, off, WAVESZ);
    if (lane == 0) psum[wave] = part;
    __syncthreads();

    if (tid == 0) {
        float s = 0.0f;
        #pragma unroll
        for (int w = 0; w < 8; ++w) s += psum[w];
        out[0] = s * (1.0f / (float)(NSLICE * BINS));
    }
}

extern "C" void kernel_launch(void* const* d_in, const int* in_sizes, int n_in,
                              void* d_out, int out_size, void* d_ws, size_t ws_size,
                              hipStream_t stream) {
    (void)in_sizes; (void)n_in; (void)out_size; (void)ws_size;
    const float* src = (const float*)d_in[0];
    const float* tgt = (const float*)d_in[1];
    // d_in[2] is bins == 256 (compile-time constant here).
    float* hist = (float*)d_ws;   // 96 * 256 floats = 96 KB of scratch

    hist_kernel<<<2 * NSLICE, THREADS, 0, stream>>>(src, tgt, hist);
    loss_kernel<<<1, 256, 0, stream>>>(hist, (float*)d_out);
}